// VSSS_66460323938611
// MI455X (gfx1250) — compile-verified
//
#include <hip/hip_runtime.h>
#include <cmath>

#define Bn 8
#define Hn 128
#define Wn 1624
#define Ln (Hn * Wn)

typedef float v2f __attribute__((ext_vector_type(2)));
typedef float v8f __attribute__((ext_vector_type(8)));

#if defined(__gfx1250__) && __has_builtin(__builtin_amdgcn_wmma_f32_16x16x4_f32)
#define USE_WMMA 1
#else
#define USE_WMMA 0
#endif

__device__ __forceinline__ float siluf(float x) { return x / (1.0f + expf(-x)); }
__device__ __forceinline__ float geluf(float x) {
  return 0.5f * x * (1.0f + erff(x * 0.70710678118654752f));
}

// ---------------------------------------------------------------------------
// Phase 1: partial channel sums over L (deterministic tree reduction, no
// float atomics). grid = (8 batches, 64 chunks), 256 threads.
// ---------------------------------------------------------------------------
__global__ __launch_bounds__(256) void vsss_reduce(
    const float* __restrict__ p0, const float* __restrict__ p1,
    const float* __restrict__ p2, const float* __restrict__ p3,
    const float* __restrict__ xin, int mode, float* __restrict__ partials) {
  const int b = blockIdx.x;
  const int chunk = blockIdx.y;
  const int per = (Ln + 63) / 64;  // 3248, 64*3248 == Ln exactly
  const int start = chunk * per;
  const int end = min(start + per, Ln);
  float s0 = 0.f, s1 = 0.f, s2 = 0.f, s3 = 0.f;
  if (mode == 0) {
    const size_t base = (size_t)b * Ln;
    for (int i = start + (int)threadIdx.x; i < end; i += 256) {
      s0 += p0[base + i]; s1 += p1[base + i];
      s2 += p2[base + i]; s3 += p3[base + i];
    }
  } else {
    const float4* xi = (const float4*)xin;
    const size_t base = (size_t)b * Ln;
    for (int i = start + (int)threadIdx.x; i < end; i += 256) {
      float4 v = xi[base + i];
      s0 += v.x; s1 += v.y; s2 += v.z; s3 += v.w;
    }
  }
  __shared__ float4 red[256];
  red[threadIdx.x] = make_float4(s0, s1, s2, s3);
  __syncthreads();
  for (int off = 128; off > 0; off >>= 1) {
    if ((int)threadIdx.x < off) {
      float4 a = red[threadIdx.x];
      float4 c = red[threadIdx.x + off];
      red[threadIdx.x] = make_float4(a.x + c.x, a.y + c.y, a.z + c.z, a.w + c.w);
    }
    __syncthreads();
  }
  if (threadIdx.x == 0) {
    float4 r = red[0];
    float* o = partials + (size_t)(b * 64 + chunk) * 4;
    o[0] = r.x; o[1] = r.y; o[2] = r.z; o[3] = r.w;
  }
}

// ---------------------------------------------------------------------------
// Phase 2: gate = sigmoid(relu(mean @ fc1^T + b1) @ fc2^T + b2).  1 block/32thr.
// ---------------------------------------------------------------------------
__global__ void vsss_gate(const float* __restrict__ partials,
                          const float* __restrict__ fc1_w, const float* __restrict__ fc1_b,
                          const float* __restrict__ fc2_w, const float* __restrict__ fc2_b,
                          float* __restrict__ gate) {
  const int t = threadIdx.x;
  if (t >= 32) return;
  const int b = t >> 2, c = t & 3;
  float m0 = 0.f, m1 = 0.f, m2 = 0.f, m3 = 0.f;
  for (int k = 0; k < 64; ++k) {
    const float* p = partials + (size_t)(b * 64 + k) * 4;
    m0 += p[0]; m1 += p[1]; m2 += p[2]; m3 += p[3];
  }
  const float invL = 1.0f / (float)Ln;
  float z = fc1_b[0] + invL * (m0 * fc1_w[0] + m1 * fc1_w[1] + m2 * fc1_w[2] + m3 * fc1_w[3]);
  z = fmaxf(z, 0.0f);
  const float g = z * fc2_w[c] + fc2_b[c];
  gate[t] = 1.0f / (1.0f + expf(-g));
}

// ---------------------------------------------------------------------------
// Phase 3: fused per-layer kernel.
//   16x16 tile + 1-pixel halo (18x18).  Stage A computes, per halo pixel:
//   degenerate ss2d -> gate -> LN -> +res -> LN -> fc1 (4->16 via
//   V_WMMA_F32_16X16X4_F32, 16 pixels per wave per op) -> LDS.
//   LDS halo plane padded to 21*16 = 336 pixel slots so all 8 D-scatter
//   stores are unconditional (boundary zeroing folds into a cndmask; no
//   divergent exec-mask branches).
//   Stage B: depthwise 3x3 conv + exact GELU + fc2 + residual, float4 store.
// ---------------------------------------------------------------------------
__global__ __launch_bounds__(256) void vsss_main(
    const float* __restrict__ p0, const float* __restrict__ p1,
    const float* __restrict__ p2, const float* __restrict__ p3,
    const float* __restrict__ xin, int mode,
    const float* __restrict__ gatev,
    const float* __restrict__ in_proj_w, const float* __restrict__ in_proj_b,
    const float* __restrict__ out_norm_b, const float* __restrict__ out_proj_w,
    const float* __restrict__ out_proj_b,
    const float* __restrict__ norm_g, const float* __restrict__ norm_b,
    const float* __restrict__ m1w, const float* __restrict__ m1b,
    const float* __restrict__ dww, const float* __restrict__ dwb,
    const float* __restrict__ m2w, const float* __restrict__ m2b,
    float* __restrict__ out) {
  __shared__ float lds_hdn[336 * 17];  // 18*18 halo (+12 pad slots) x 16 ch, pad 17
  __shared__ float lds_xm2[256 * 4];   // interior residual values

  const int tid = threadIdx.x;
  const int lane = tid & 31;
  const int wav = tid >> 5;
  const int half = lane >> 4;
  const int n16 = lane & 15;

  const int gx0 = blockIdx.x * 16;
  const int gy0 = blockIdx.y * 16;
  const int b = blockIdx.z;

  // ss2d degenerates to: c1*silu(x*az+bz)+c2 (LN over singleton axis == bias)
  const float az = in_proj_w[1];
  const float bz = in_proj_b[1];
  const float c1 = out_norm_b[0] * out_proj_w[0];
  const float c2 = out_proj_b[0];
  const float g0 = gatev[b * 4 + 0], g1 = gatev[b * 4 + 1];
  const float g2 = gatev[b * 4 + 2], g3 = gatev[b * 4 + 3];
  const float ng0 = norm_g[0], ng1 = norm_g[1], ng2 = norm_g[2], ng3 = norm_g[3];
  const float nb0 = norm_b[0], nb1 = norm_b[1], nb2 = norm_b[2], nb3 = norm_b[3];

#if USE_WMMA
  // B operand: B[k][n] = mlp_fc1_w[n][k].  Lane n16 holds column n16;
  // half 0 -> K{0,1}, half 1 -> K{2,3} (mirrors the A-matrix striping).
  v2f bmat;
  bmat.x = m1w[n16 * 4 + half * 2 + 0];
  bmat.y = m1w[n16 * 4 + half * 2 + 1];
  const float fc1bias = m1b[n16];
#endif

  // ---- Stage A over 21 groups of 16 halo pixels (8 waves x up to 3) ----
  for (int it = 0; it < 3; ++it) {
    const int g = wav + it * 8;
    if (g >= 21) break;  // uniform per wave
    const int hidx = g * 16 + n16;
    const int hidxc = hidx < 324 ? hidx : 323;
    const int hy = hidxc / 18;
    const int wx = hidxc - hy * 18;
    const int gy = gy0 - 1 + hy;
    const int gx = gx0 - 1 + wx;
    const bool img = (hidx < 324) & ((unsigned)gy < (unsigned)Hn) & ((unsigned)gx < (unsigned)Wn);
    const int gyc = min(max(gy, 0), Hn - 1);
    const int gxc = min(max(gx, 0), Wn - 1);
    float x0, x1, x2, x3;
    if (mode == 0) {
      const size_t off = (size_t)b * Ln + (size_t)gyc * Wn + gxc;
      x0 = p0[off]; x1 = p1[off]; x2 = p2[off]; x3 = p3[off];
    } else {
      const float4 v = ((const float4*)xin)[(size_t)b * Ln + (size_t)gyc * Wn + gxc];
      x0 = v.x; x1 = v.y; x2 = v.z; x3 = v.w;
    }
    if (!img) { x0 = 0.f; x1 = 0.f; x2 = 0.f; x3 = 0.f; }

    // degenerate ss2d per channel, times gate
    float a0 = (c1 * siluf(x0 * az + bz) + c2) * g0;
    float a1 = (c1 * siluf(x1 * az + bz) + c2) * g1;
    float a2 = (c1 * siluf(x2 * az + bz) + c2) * g2;
    float a3 = (c1 * siluf(x3 * az + bz) + c2) * g3;
    // LN over 4 channels, then residual
    float mu = 0.25f * (a0 + a1 + a2 + a3);
    float d0 = a0 - mu, d1 = a1 - mu, d2 = a2 - mu, d3 = a3 - mu;
    float rs = rsqrtf(0.25f * (d0 * d0 + d1 * d1 + d2 * d2 + d3 * d3) + 1e-5f);
    float y0 = x0 + d0 * rs * ng0 + nb0;
    float y1 = x1 + d1 * rs * ng1 + nb1;
    float y2 = x2 + d2 * rs * ng2 + nb2;
    float y3 = x3 + d3 * rs * ng3 + nb3;
    // second LN feeding the MLP
    float mu2 = 0.25f * (y0 + y1 + y2 + y3);
    float e0 = y0 - mu2, e1 = y1 - mu2, e2 = y2 - mu2, e3 = y3 - mu2;
    float rs2 = rsqrtf(0.25f * (e0 * e0 + e1 * e1 + e2 * e2 + e3 * e3) + 1e-5f);
    float t0 = e0 * rs2 * ng0 + nb0;
    float t1 = e1 * rs2 * ng1 + nb1;
    float t2 = e2 * rs2 * ng2 + nb2;
    float t3 = e3 * rs2 * ng3 + nb3;

    // stash residual for interior pixels (half 0 lanes own the pixel)
    if (half == 0 && hidx < 324 && hy >= 1 && hy <= 16 && wx >= 1 && wx <= 16) {
      const int ii = (hy - 1) * 16 + (wx - 1);
      lds_xm2[ii * 4 + 0] = y0; lds_xm2[ii * 4 + 1] = y1;
      lds_xm2[ii * 4 + 2] = y2; lds_xm2[ii * 4 + 3] = y3;
    }

#if USE_WMMA
    // A operand: lane holds row M = n16 (its pixel), K pair by half.
    v2f amat;
    amat.x = half ? t2 : t0;
    amat.y = half ? t3 : t1;
    v8f acc = {};
    acc = __builtin_amdgcn_wmma_f32_16x16x4_f32(
        false, amat, false, bmat, (short)0, acc, false, false);
    // D: VGPR r holds pixel M = r + 8*half, column N = n16 (hidden channel).
    // Unconditional scatter into the padded 336-slot plane; out-of-image /
    // out-of-range slots get 0 via select (zero 'SAME' padding for the conv).
#pragma unroll
    for (int r = 0; r < 8; ++r) {
      const int h2 = g * 16 + r + half * 8;  // < 336 always
      const int hy2 = h2 / 18;
      const int wx2 = h2 - hy2 * 18;
      const int gy2 = gy0 - 1 + hy2;
      const int gx2 = gx0 - 1 + wx2;
      const bool img2 = (h2 < 324) & ((unsigned)gy2 < (unsigned)Hn) &
                        ((unsigned)gx2 < (unsigned)Wn);
      lds_hdn[h2 * 17 + n16] = img2 ? (acc[r] + fc1bias) : 0.0f;
    }
#else
    if (half == 0 && hidx < 324) {
#pragma unroll
      for (int n = 0; n < 16; ++n) {
        float h = m1b[n] + t0 * m1w[n * 4 + 0] + t1 * m1w[n * 4 + 1] +
                  t2 * m1w[n * 4 + 2] + t3 * m1w[n * 4 + 3];
        lds_hdn[hidx * 17 + n] = img ? h : 0.0f;
      }
    }
#endif
  }

  __syncthreads();

  // ---- Stage B: depthwise conv3x3 (zero 'SAME' pad) + GELU + fc2 + res ----
  const int py = tid >> 4, px = tid & 15;
  const int gy = gy0 + py, gx = gx0 + px;
  if (gx < Wn) {
    float o0 = m2b[0], o1 = m2b[1], o2 = m2b[2], o3 = m2b[3];
#pragma unroll
    for (int c = 0; c < 16; ++c) {
      float s = dwb[c];
#pragma unroll
      for (int ky = 0; ky < 3; ++ky) {
#pragma unroll
        for (int kx = 0; kx < 3; ++kx) {
          s += lds_hdn[((py + ky) * 18 + (px + kx)) * 17 + c] *
               dww[(ky * 3 + kx) * 16 + c];
        }
      }
      s = geluf(s);
      o0 += s * m2w[0 * 16 + c];
      o1 += s * m2w[1 * 16 + c];
      o2 += s * m2w[2 * 16 + c];
      o3 += s * m2w[3 * 16 + c];
    }
    const int ii = py * 16 + px;
    float4 r;
    r.x = lds_xm2[ii * 4 + 0] + o0;
    r.y = lds_xm2[ii * 4 + 1] + o1;
    r.z = lds_xm2[ii * 4 + 2] + o2;
    r.w = lds_xm2[ii * 4 + 3] + o3;
    ((float4*)out)[(size_t)b * Ln + (size_t)gy * Wn + gx] = r;
  }
}

// ---------------------------------------------------------------------------
extern "C" void kernel_launch(void* const* d_in, const int* in_sizes, int n_in,
                              void* d_out, int out_size, void* d_ws, size_t ws_size,
                              hipStream_t stream) {
  const float* w1 = (const float*)d_in[0];
  const float* w2 = (const float*)d_in[1];
  const float* w3 = (const float*)d_in[2];
  const float* w4 = (const float*)d_in[3];
  const float* fc1_w = (const float*)d_in[4];
  const float* fc1_b = (const float*)d_in[5];
  const float* fc2_w = (const float*)d_in[6];
  const float* fc2_b = (const float*)d_in[7];
  const float* norm_g = (const float*)d_in[8];
  const float* norm_b = (const float*)d_in[9];
  const float* m1w = (const float*)d_in[10];
  const float* m1b = (const float*)d_in[11];
  const float* dww = (const float*)d_in[12];
  const float* dwb = (const float*)d_in[13];
  const float* m2w = (const float*)d_in[14];
  const float* m2b = (const float*)d_in[15];
  const float* in_proj_w = (const float*)d_in[16];
  const float* in_proj_b = (const float*)d_in[17];
  const float* out_norm_b = (const float*)d_in[26];
  const float* out_proj_w = (const float*)d_in[27];
  const float* out_proj_b = (const float*)d_in[28];

  float* ws = (float*)d_ws;
  float* ws_gate = ws;         // 32 floats
  float* ws_part = ws + 32;    // 8*64*4 = 2048 floats
  float* ws_inter = ws + 4096; // B*L*4 floats (16B aligned), layer-1 output

  const dim3 gridR(Bn, 64), blockR(256);
  const dim3 gridM((Wn + 15) / 16, Hn / 16, Bn), blockM(256);

  // ---- layer 1: planar inputs -> ws_inter ----
  vsss_reduce<<<gridR, blockR, 0, stream>>>(w1, w2, w3, w4, nullptr, 0, ws_part);
  vsss_gate<<<1, 32, 0, stream>>>(ws_part, fc1_w, fc1_b, fc2_w, fc2_b, ws_gate);
  vsss_main<<<gridM, blockM, 0, stream>>>(
      w1, w2, w3, w4, nullptr, 0, ws_gate, in_proj_w, in_proj_b,
      out_norm_b, out_proj_w, out_proj_b, norm_g, norm_b,
      m1w, m1b, dww, dwb, m2w, m2b, ws_inter);

  // ---- layer 2: interleaved ws_inter -> d_out ----
  vsss_reduce<<<gridR, blockR, 0, stream>>>(nullptr, nullptr, nullptr, nullptr,
                                            ws_inter, 1, ws_part);
  vsss_gate<<<1, 32, 0, stream>>>(ws_part, fc1_w, fc1_b, fc2_w, fc2_b, ws_gate);
  vsss_main<<<gridM, blockM, 0, stream>>>(
      nullptr, nullptr, nullptr, nullptr, ws_inter, 1, ws_gate,
      in_proj_w, in_proj_b, out_norm_b, out_proj_w, out_proj_b, norm_g, norm_b,
      m1w, m1b, dww, dwb, m2w, m2b, (float*)d_out);
}